// MultiHeadedSelfAttention_429496729636
// MI455X (gfx1250) — compile-verified
//
#include <hip/hip_runtime.h>
#include <hip/hip_bf16.h>

typedef __attribute__((ext_vector_type(2))) float v2f;
typedef __attribute__((ext_vector_type(8))) float v8f;

#define B_ 4
#define S_ 2048
#define D_ 1024
#define H_ 16
#define W_ 64
#define LDSS 68   // 64 + 4 pad: 16B-aligned float4 LDS ops, bank spread
#define LDSB 132  // 128 + 4 pad for the wide B tile

__device__ __forceinline__ v8f wmma4(v2f a, v2f b, v8f c) {
  // V_WMMA_F32_16X16X4_F32: D = A(16x4,f32) * B(4x16,f32) + C(16x16,f32)
  return __builtin_amdgcn_wmma_f32_16x16x4_f32(false, a, false, b, (short)0, c,
                                               false, false);
}

// Stage a 64x64 fp32 tile (row-major, src_stride floats) into LDS (stride LDSS).
__device__ __forceinline__ void load_tile64(const float* __restrict__ src,
                                            int src_stride, float* dst, int tid) {
#pragma unroll
  for (int i = 0; i < 4; ++i) {
    int idx = tid + i * 256;       // 1024 float4s
    int row = idx >> 4;
    int c4  = (idx & 15) << 2;
    float4 t = *(const float4*)(src + row * src_stride + c4);
    *(float4*)(dst + row * LDSS + c4) = t;
  }
}

// Stage a 64x128 fp32 tile into LDS (stride LDSB).
__device__ __forceinline__ void load_tile128(const float* __restrict__ src,
                                             int src_stride, float* dst, int tid) {
#pragma unroll
  for (int i = 0; i < 8; ++i) {
    int idx = tid + i * 256;       // 2048 float4s
    int row = idx >> 5;            // 32 float4 per row
    int c4  = (idx & 31) << 2;
    float4 t = *(const float4*)(src + row * src_stride + c4);
    *(float4*)(dst + row * LDSB + c4) = t;
  }
}

// Prefetch a 64-row tile (row_floats floats per row) into L2/WGP$ (global_prefetch_b8).
__device__ __forceinline__ void prefetch_tile(const float* src, int src_stride,
                                              int row_floats, int tid) {
  // one prefetch per 128B cacheline: rows*ceil(row_floats/32) lines
  int lines_per_row = row_floats >> 5;
  int nlines = 64 * lines_per_row;
  for (int l = tid; l < nlines; l += 256) {
    int row = l / lines_per_row;
    int seg = l % lines_per_row;
    __builtin_prefetch(src + row * src_stride + seg * 32, 0, 3);
  }
}

// ---------------------------------------------------------------------------
// Kernel 1: y = x @ W + b for W in {Wq,Wk,Wv}; scatter into (B,H,S,W) layout.
// Block = 256 threads (8 waves). Block tile 64(M) x 128(N).
// Wave tile 16(M) x 64(N): A fragment reused across 4 WMMAs.
// ---------------------------------------------------------------------------
__global__ __launch_bounds__(256) void qkv_proj_kernel(
    const float* __restrict__ x,
    const float* __restrict__ Wq, const float* __restrict__ bq,
    const float* __restrict__ Wk, const float* __restrict__ bk,
    const float* __restrict__ Wv, const float* __restrict__ bv,
    float* __restrict__ qo, float* __restrict__ ko, float* __restrict__ vo) {
  __shared__ float As[64 * LDSS];   // x tile: 64 rows x 64 k
  __shared__ float Bs[64 * LDSB];   // W tile: 64 k x 128 n

  const int tid  = threadIdx.x;
  const int lane = tid & 31;
  const int ll   = lane & 15;
  const int lh   = lane >> 4;
  const int wid  = tid >> 5;
  const int mi   = wid & 3;     // 16-row subtile (0..3)
  const int nj   = wid >> 2;    // 64-col half (0..1)

  const int nbase = blockIdx.x * 128;
  const int mbase = blockIdx.y * 64;
  const int which = blockIdx.z;
  const float* Wm = which == 0 ? Wq : (which == 1 ? Wk : Wv);
  const float* bm = which == 0 ? bq : (which == 1 ? bk : bv);
  float*       dst = which == 0 ? qo : (which == 1 ? ko : vo);

  v8f acc[4] = {v8f{}, v8f{}, v8f{}, v8f{}};

  for (int kt = 0; kt < D_ / 64; ++kt) {
    load_tile64(x + mbase * D_ + kt * 64, D_, As, tid);
    load_tile128(Wm + (kt * 64) * D_ + nbase, D_, Bs, tid);
    if (kt + 1 < D_ / 64) {   // warm L2/WGP$ for next k-tile while we compute
      prefetch_tile(x + mbase * D_ + (kt + 1) * 64, D_, 64, tid);
      prefetch_tile(Wm + ((kt + 1) * 64) * D_ + nbase, D_, 128, tid);
    }
    __syncthreads();
#pragma unroll
    for (int kk = 0; kk < 16; ++kk) {
      const int kb = kk * 4 + lh * 2;
      v2f a;
      a.x = As[(mi * 16 + ll) * LDSS + kb + 0];
      a.y = As[(mi * 16 + ll) * LDSS + kb + 1];
#pragma unroll
      for (int t = 0; t < 4; ++t) {
        v2f b;
        b.x = Bs[(kb + 0) * LDSB + nj * 64 + t * 16 + ll];
        b.y = Bs[(kb + 1) * LDSB + nj * 64 + t * 16 + ll];
        acc[t] = wmma4(a, b, acc[t]);
      }
    }
    __syncthreads();
  }

#pragma unroll
  for (int t = 0; t < 4; ++t) {
#pragma unroll
    for (int vv = 0; vv < 8; ++vv) {
      int gr = mbase + mi * 16 + lh * 8 + vv;   // global row in (B*S)
      int bb = gr >> 11;                        // / S_
      int ss = gr & (S_ - 1);
      int gc = nbase + nj * 64 + t * 16 + ll;
      int hh = gc >> 6, ww = gc & 63;
      dst[(((bb * H_ + hh) * S_) + ss) * W_ + ww] = acc[t][vv] + bm[gc];
    }
  }
}

// ---------------------------------------------------------------------------
// Kernel 2: flash attention. Block = (b, h, 64 query rows); online softmax.
// Ss buffer holds the K tile, then (after a barrier) the P=exp(S) tile.
// ---------------------------------------------------------------------------
__global__ __launch_bounds__(256) void flash_attn_kernel(
    const float* __restrict__ q, const float* __restrict__ k,
    const float* __restrict__ v, const int* __restrict__ mask,
    float* __restrict__ out) {
  __shared__ float Qs[64 * LDSS];
  __shared__ float Ss[64 * LDSS];
  __shared__ float Vs[64 * LDSS];
  __shared__ float part[64][4];
  __shared__ float row_m[64], row_l[64], row_f[64];
  __shared__ float mskf[64];

  const int tid  = threadIdx.x;
  const int lane = tid & 31;
  const int ll   = lane & 15;
  const int lh   = lane >> 4;
  const int wid  = tid >> 5;
  const int mi   = wid & 3;
  const int nj   = wid >> 2;

  const int rt = blockIdx.x, hh = blockIdx.y, bb = blockIdx.z;
  const int qrow0 = rt * 64;
  const float* qh = q + (size_t)((bb * H_ + hh) * S_) * W_;
  const float* kh = k + (size_t)((bb * H_ + hh) * S_) * W_;
  const float* vh = v + (size_t)((bb * H_ + hh) * S_) * W_;

  load_tile64(qh + qrow0 * W_, W_, Qs, tid);
  if (tid < 64) { row_m[tid] = -1e30f; row_l[tid] = 0.0f; }

  v8f acc0 = {}; v8f acc1 = {};

  for (int kt = 0; kt < S_ / 64; ++kt) {
    load_tile64(kh + kt * 64 * W_, W_, Ss, tid);   // K tile into Ss
    load_tile64(vh + kt * 64 * W_, W_, Vs, tid);
    if (tid < 64) mskf[tid] = 10000.0f * (float)mask[bb * S_ + kt * 64 + tid];
    if (kt + 1 < S_ / 64) {   // overlap next tile fetch with this tile's math
      prefetch_tile(kh + (kt + 1) * 64 * W_, W_, 64, tid);
      prefetch_tile(vh + (kt + 1) * 64 * W_, W_, 64, tid);
    }
    __syncthreads();

    // S = Q @ K^T  (A = Q[m][w], B[w][n] = K[n][w])
    v8f s0 = {}; v8f s1 = {};
#pragma unroll
    for (int kk = 0; kk < 16; ++kk) {
      const int kb = kk * 4 + lh * 2;
      v2f a, b0, b1;
      a.x  = Qs[(mi * 16 + ll) * LDSS + kb + 0];
      a.y  = Qs[(mi * 16 + ll) * LDSS + kb + 1];
      b0.x = Ss[(nj * 32 + ll) * LDSS + kb + 0];
      b0.y = Ss[(nj * 32 + ll) * LDSS + kb + 1];
      b1.x = Ss[(nj * 32 + 16 + ll) * LDSS + kb + 0];
      b1.y = Ss[(nj * 32 + 16 + ll) * LDSS + kb + 1];
      s0 = wmma4(a, b0, s0);
      s1 = wmma4(a, b1, s1);
    }
    __syncthreads();  // all waves done reading K tile

    // scale + mask, write scores into Ss (C layout -> LDS rows)
#pragma unroll
    for (int vv = 0; vv < 8; ++vv) {
      int r  = mi * 16 + lh * 8 + vv;
      int c0 = nj * 32 + ll, c1 = c0 + 16;
      Ss[r * LDSS + c0] = s0[vv] * 0.125f - mskf[c0];
      Ss[r * LDSS + c1] = s1[vv] * 0.125f - mskf[c1];
    }
    __syncthreads();

    // per-row max of this tile (4 threads per row)
    {
      int r = tid >> 2, sg = tid & 3;
      float mx = -1e30f;
#pragma unroll
      for (int c = 0; c < 16; ++c)
        mx = fmaxf(mx, Ss[r * LDSS + sg * 16 + c]);
      part[r][sg] = mx;
    }
    __syncthreads();
    if (tid < 64) {
      float tm = fmaxf(fmaxf(part[tid][0], part[tid][1]),
                       fmaxf(part[tid][2], part[tid][3]));
      float m_new = fmaxf(row_m[tid], tm);
      float f = __expf(row_m[tid] - m_new);
      row_f[tid] = f;
      row_m[tid] = m_new;
      row_l[tid] *= f;
    }
    __syncthreads();

    // P = exp(S - m); partial sums; rescale O accumulators by row_f
    {
      int r = tid >> 2, sg = tid & 3;
      float m_new = row_m[r];
      float sum = 0.0f;
#pragma unroll
      for (int c = 0; c < 16; ++c) {
        float e = __expf(Ss[r * LDSS + sg * 16 + c] - m_new);
        Ss[r * LDSS + sg * 16 + c] = e;
        sum += e;
      }
      part[r][sg] = sum;
    }
#pragma unroll
    for (int vv = 0; vv < 8; ++vv) {
      float f = row_f[mi * 16 + lh * 8 + vv];
      acc0[vv] *= f;
      acc1[vv] *= f;
    }
    __syncthreads();
    if (tid < 64)
      row_l[tid] += part[tid][0] + part[tid][1] + part[tid][2] + part[tid][3];

    // O += P @ V
#pragma unroll
    for (int kk = 0; kk < 16; ++kk) {
      const int kb = kk * 4 + lh * 2;
      v2f a, b0, b1;
      a.x  = Ss[(mi * 16 + ll) * LDSS + kb + 0];
      a.y  = Ss[(mi * 16 + ll) * LDSS + kb + 1];
      b0.x = Vs[(kb + 0) * LDSS + nj * 32 + ll];
      b0.y = Vs[(kb + 1) * LDSS + nj * 32 + ll];
      b1.x = Vs[(kb + 0) * LDSS + nj * 32 + 16 + ll];
      b1.y = Vs[(kb + 1) * LDSS + nj * 32 + 16 + ll];
      acc0 = wmma4(a, b0, acc0);
      acc1 = wmma4(a, b1, acc1);
    }
    __syncthreads();  // protect Ss/Vs/part before next tile
  }

  // normalize and write (B,S,D)
#pragma unroll
  for (int vv = 0; vv < 8; ++vv) {
    int r = mi * 16 + lh * 8 + vv;
    float inv = 1.0f / row_l[r];
    int sg = qrow0 + r;
    int c0 = nj * 32 + ll, c1 = c0 + 16;
    out[(size_t)(bb * S_ + sg) * D_ + hh * W_ + c0] = acc0[vv] * inv;
    out[(size_t)(bb * S_ + sg) * D_ + hh * W_ + c1] = acc1[vv] * inv;
  }
}

extern "C" void kernel_launch(void* const* d_in, const int* in_sizes, int n_in,
                              void* d_out, int out_size, void* d_ws, size_t ws_size,
                              hipStream_t stream) {
  (void)in_sizes; (void)n_in; (void)out_size; (void)ws_size;
  const float* x  = (const float*)d_in[0];
  const float* Wq = (const float*)d_in[1];
  const float* bq = (const float*)d_in[2];
  const float* Wk = (const float*)d_in[3];
  const float* bk = (const float*)d_in[4];
  const float* Wv = (const float*)d_in[5];
  const float* bv = (const float*)d_in[6];
  const int*  msk = (const int*)d_in[7];
  float* out = (float*)d_out;

  // workspace: Q,K,V in (B,H,S,W) fp32 — 3 * 32 MiB = 96 MiB
  float* qws = (float*)d_ws;
  float* kws = qws + (size_t)B_ * S_ * D_;
  float* vws = kws + (size_t)B_ * S_ * D_;

  dim3 g1(D_ / 128, (B_ * S_) / 64, 3);
  qkv_proj_kernel<<<g1, 256, 0, stream>>>(x, Wq, bq, Wk, bk, Wv, bv,
                                          qws, kws, vws);

  dim3 g2(S_ / 64, H_, B_);
  flash_attn_kernel<<<g2, 256, 0, stream>>>(qws, kws, vws, msk, out);
}